// HierarchicalHashEmbedding_42863773614124
// MI455X (gfx1250) — compile-verified
//
#include <hip/hip_runtime.h>

// HierarchicalHashEmbedding forward (all-hit path) == batched gather:
//   out[row, :] = table[indices[row], :]   row in [0, B*F), D = 64
//
// Memory-bound: ~420 MB traffic -> ~18 us floor at 23.3 TB/s HBM.
//  - 4-wide batched 128-bit gathers per wave iteration: 4 independent
//    global_load_b128 in flight (2 KB read + 2 KB write per wave iter)
//    so per-wave MLP, not just occupancy, hides random-row HBM latency.
//  - Table loads temporal: ~559K distinct rows ~= 143 MB < 192 MB L2.
//  - Output stores non-temporal (write-once 210 MB stream, keep L2 for table).
//  - One clamped speculative global_prefetch_b8 per iteration for the next
//    chunk's random row (OOB-safe, branch-free).

typedef float v4f __attribute__((ext_vector_type(4)));

namespace {
constexpr int kVecPerRow = 16;    // 64 floats / 4 per float4
constexpr int kThreads   = 256;   // 8 wave32 per block
constexpr int kBlocks    = 4096;  // ~1M threads
}

__global__ __launch_bounds__(kThreads) void
hier_embed_gather(const int* __restrict__ indices,
                  const float* __restrict__ table,
                  float* __restrict__ out,
                  int nrows) {
  const int tid  = blockIdx.x * blockDim.x + threadIdx.x;
  const int step = gridDim.x * blockDim.x;
  const int nvec = nrows * kVecPerRow;   // 13,107,200 for reference shape

  const v4f* __restrict__ tab4 = (const v4f*)table;
  v4f* __restrict__       out4 = (v4f*)out;

  int v = tid;

  // Main loop: 4 independent gathers in flight per wave (MLP = 4).
  for (; v + 3 * step < nvec; v += 4 * step) {
    const int v0 = v, v1 = v + step, v2 = v + 2 * step, v3 = v + 3 * step;

    // Index dwords (lane-groups share addresses -> coalesced b32 loads).
    const int e0 = indices[v0 >> 4];
    const int e1 = indices[v1 >> 4];
    const int e2 = indices[v2 >> 4];
    const int e3 = indices[v3 >> 4];

    // Speculative prefetch of the NEXT chunk's first random row; the index
    // read is clamped in-bounds, the prefetch itself is OOB-safe.
    const int vp  = v + 4 * step;
    const int vpc = (vp < nvec) ? vp : v;
    const int ep  = indices[vpc >> 4];
    __builtin_prefetch(&tab4[ep * kVecPerRow + (vp & 15)], /*rw=*/0, /*loc=*/1);

    // Four independent 128-bit gather loads (temporal: L2 reuse of dup rows).
    const v4f a = tab4[e0 * kVecPerRow + (v0 & 15)];
    const v4f b = tab4[e1 * kVecPerRow + (v1 & 15)];
    const v4f c = tab4[e2 * kVecPerRow + (v2 & 15)];
    const v4f d = tab4[e3 * kVecPerRow + (v3 & 15)];

    // Non-temporal streaming stores; out is contiguous so dst == out4 + v.
    __builtin_nontemporal_store(a, &out4[v0]);
    __builtin_nontemporal_store(b, &out4[v1]);
    __builtin_nontemporal_store(c, &out4[v2]);
    __builtin_nontemporal_store(d, &out4[v3]);
  }

  // Tail: remaining elements one at a time.
  for (; v < nvec; v += step) {
    const int e   = indices[v >> 4];
    const v4f val = tab4[e * kVecPerRow + (v & 15)];
    __builtin_nontemporal_store(val, &out4[v]);
  }
}

extern "C" void kernel_launch(void* const* d_in, const int* in_sizes, int n_in,
                              void* d_out, int out_size, void* d_ws, size_t ws_size,
                              hipStream_t stream) {
  const int*   indices = (const int*)d_in[0];   // [B, F] int32
  const float* table   = (const float*)d_in[1]; // [V, D] float32
  float*       out     = (float*)d_out;         // [B, F, D] float32

  const int nrows = in_sizes[0];                // B*F = 819,200
  const int nvec  = nrows * kVecPerRow;
  int blocks = (nvec + kThreads - 1) / kThreads;
  if (blocks > kBlocks) blocks = kBlocks;

  hier_embed_gather<<<blocks, kThreads, 0, stream>>>(indices, table, out, nrows);
}